// BoundaryPredictorMLP_43808666419254
// MI455X (gfx1250) — compile-verified
//
#include <hip/hip_runtime.h>
#include <hip/hip_bf16.h>
#include <math.h>

typedef float v2f __attribute__((ext_vector_type(2)));
typedef float v8f __attribute__((ext_vector_type(8)));

#define BDIM 8
#define SDIM 2048
#define DDIM 768
#define MTOT (BDIM * SDIM)   // 16384

// ---------------- GEMM tile config ----------------
#define BM 128
#define BN 128
#define BK 32
#define ASTRIDE (BK + 4)     // 36 floats; frag pair (m*36 + c) is 8B-aligned
#define PBSTRIDE (BN + 2)    // 130 float2 per K-pair row; 16B-aligned float4 slots

__global__ void zero_kernel(float* __restrict__ logits, int* __restrict__ nb) {
    int i = blockIdx.x * 256 + threadIdx.x;
    if (i < MTOT) logits[i] = 0.0f;
    if (i < BDIM) nb[i] = 0;
}

// logits[m] += sum_n gelu(hidden_row_m . W1_col_n + b1[n]) * W2[n]
__global__ __launch_bounds__(256) void gemm_gelu_dot(
    const float* __restrict__ hidden,   // [MTOT, DDIM] row-major
    const float* __restrict__ W1,       // [DDIM, DDIM] row-major (K x N)
    const float* __restrict__ b1,       // [DDIM]
    const float* __restrict__ W2,       // [DDIM]
    float* __restrict__ logits)         // [MTOT], accumulated with atomics
{
    __shared__ float  As[BM * ASTRIDE];
    __shared__ float2 Bs[(BK / 2) * PBSTRIDE];   // {W1[k][n], W1[k+1][n]} pairs

    const int tid   = threadIdx.x;
    const int lane  = tid & 31;
    const int wave  = tid >> 5;
    const int l15   = lane & 15;
    const int half  = lane >> 4;        // 0: lanes 0-15 (K=c,c+1), 1: lanes 16-31 (K=c+2,c+3)
    const int mBase = blockIdx.x * BM;
    const int nBase = blockIdx.y * BN;
    const int waveM = (wave & 3) * 32;  // 4 waves tile M (32 rows each)
    const int waveN = (wave >> 2) * 64; // 2 waves tile N (64 cols each)

    v8f acc[2][4];
#pragma unroll
    for (int i = 0; i < 2; ++i)
#pragma unroll
        for (int j = 0; j < 4; ++j)
#pragma unroll
            for (int r = 0; r < 8; ++r) acc[i][j][r] = 0.0f;

    // A global->LDS: 128x32 floats, 16 per thread (4x float4)
    const int ar = tid >> 3;            // row group 0..31
    const int aq = (tid & 7) * 4;       // col quad within BK
    // B global->LDS: 32x128 floats as 2x2 quads; 16 pair-rows x 64 col-pairs
    const int qcol  = tid & 63;         // col-pair 0..63
    const int prow0 = tid >> 6;         // pair-row base 0..3

    for (int k0 = 0; k0 < DDIM; k0 += BK) {
#pragma unroll
        for (int i = 0; i < 4; ++i) {
            int row = ar + i * 32;
            float4 v = *(const float4*)(hidden + (size_t)(mBase + row) * DDIM + k0 + aq);
            *(float4*)&As[row * ASTRIDE + aq] = v;
        }
#pragma unroll
        for (int i = 0; i < 4; ++i) {
            int p = prow0 + i * 4;                 // K-pair row 0..15
            int k = k0 + p * 2;
            int n = nBase + qcol * 2;
            float2 r0 = *(const float2*)(W1 + (size_t)k * DDIM + n);
            float2 r1 = *(const float2*)(W1 + (size_t)(k + 1) * DDIM + n);
            float4 st;                             // K-pair interleave
            st.x = r0.x; st.y = r1.x; st.z = r0.y; st.w = r1.y;
            *(float4*)&Bs[p * PBSTRIDE + qcol * 2] = st;
        }
        __syncthreads();

#pragma unroll
        for (int kk = 0; kk < BK; kk += 4) {
            const int c = kk + half * 2;
            v2f aF[2], bF[4];
#pragma unroll
            for (int i = 0; i < 2; ++i) {
                int m = waveM + i * 16 + l15;
                aF[i] = *(const v2f*)&As[m * ASTRIDE + c];       // single b64
            }
#pragma unroll
            for (int j = 0; j < 4; ++j) {
                int n = waveN + j * 16 + l15;
                bF[j] = *(const v2f*)&Bs[(c >> 1) * PBSTRIDE + n]; // single b64
            }
#pragma unroll
            for (int i = 0; i < 2; ++i)
#pragma unroll
                for (int j = 0; j < 4; ++j)
                    acc[i][j] = __builtin_amdgcn_wmma_f32_16x16x4_f32(
                        false, aF[i], false, bF[j], (short)0, acc[i][j], false, false);
        }
        __syncthreads();
    }

    // Epilogue: x = acc + b1[n]; h = gelu(x); contrib = h * W2[n]; reduce over N.
#pragma unroll
    for (int i = 0; i < 2; ++i) {
#pragma unroll
        for (int r = 0; r < 8; ++r) {
            float partial = 0.0f;
#pragma unroll
            for (int j = 0; j < 4; ++j) {
                int n = nBase + waveN + j * 16 + l15;
                float x = acc[i][j][r] + b1[n];
                float h = 0.5f * x * (1.0f + erff(x * 0.70710678118654752f));
                partial += h * W2[n];
            }
#pragma unroll
            for (int off = 8; off > 0; off >>= 1)
                partial += __shfl_xor(partial, off, 16);
            if (l15 == 0) {
                int m = mBase + waveM + i * 16 + r + half * 8;
                atomicAdd(&logits[m], partial);
            }
        }
    }
}

__global__ void boundary_kernel(const float* __restrict__ logits,
                                const float* __restrict__ lengths,
                                const float* __restrict__ b2,
                                float* __restrict__ masked_probs,  // [B,S] output slice
                                float* __restrict__ hard_ws,       // [B,S]
                                int* __restrict__ nb_ws)           // [B]
{
    int m = blockIdx.x * 256 + threadIdx.x;
    if (m >= MTOT) return;
    int b = m >> 11, s = m & (SDIM - 1);
    int alen = (int)(lengths[b] * (float)SDIM);     // trunc, matches astype(int32)
    int last = min(max(alen - 1, 0), SDIM - 1);
    float x = logits[m] + b2[0];
    float p = 1.0f / (1.0f + expf(-x));
    bool valid = s < alen;
    masked_probs[m] = valid ? p : 0.0f;
    float hard = (valid && p > 0.5f) ? 1.0f : 0.0f;
    if (s == last) hard = 1.0f;                     // forced boundary at last valid
    hard_ws[m] = hard;
    if (hard != 0.0f && valid) atomicAdd(&nb_ws[b], 1);
}

__global__ void finalize_kernel(const int* __restrict__ nb_ws,
                                const float* __restrict__ lengths,
                                float* __restrict__ out_sh,
                                float* __restrict__ out_nb,
                                float* __restrict__ out_al,
                                int max_segments)
{
    int b = threadIdx.x;
    if (b >= BDIM) return;
    float nb = (float)nb_ws[b];
    float Kf = (float)max_segments;
    float sh;
    if (nb >= Kf - 1.0f)      sh = 1.0f;
    else if (nb > 0.0f)       sh = (nb + 1.0f) / Kf;
    else                      sh = 0.0f;
    out_sh[b] = sh;
    out_nb[b] = nb;
    out_al[b] = (float)((int)(lengths[b] * (float)SDIM));
}

// Segmented mean over S: one block per (batch, 128-wide D chunk).
__global__ __launch_bounds__(128) void pool_kernel(
    const float* __restrict__ hidden,
    const float* __restrict__ hard_ws,
    const float* __restrict__ lengths,
    float* __restrict__ pooled,   // [B, K, D]
    int K)
{
    __shared__ float sh_hard[SDIM];
    const int b = blockIdx.x;
    const int d = blockIdx.y * 128 + threadIdx.x;
    for (int s = threadIdx.x; s < SDIM; s += 128)
        sh_hard[s] = hard_ws[b * SDIM + s];
    __syncthreads();

    int alen = (int)(lengths[b] * (float)SDIM);
    if (alen > SDIM) alen = SDIM;

    const float* hrow = hidden + (size_t)b * SDIM * DDIM + d;
    float acc = 0.0f;
    int cnt = 0, k = 0;
    for (int s = 0; s < alen; ++s) {
        acc += hrow[(size_t)s * DDIM];
        ++cnt;
        if (sh_hard[s] != 0.0f) {
            if (k < K)
                pooled[((size_t)b * K + k) * DDIM + d] = acc / (float)cnt;
            ++k; acc = 0.0f; cnt = 0;
        }
    }
    for (; k < K; ++k)
        pooled[((size_t)b * K + k) * DDIM + d] = 0.0f;
}

extern "C" void kernel_launch(void* const* d_in, const int* in_sizes, int n_in,
                              void* d_out, int out_size, void* d_ws, size_t ws_size,
                              hipStream_t stream) {
    const float* hidden  = (const float*)d_in[0];
    const float* lengths = (const float*)d_in[1];
    const float* W1      = (const float*)d_in[2];
    const float* b1      = (const float*)d_in[3];
    const float* W2      = (const float*)d_in[4];
    const float* b2      = (const float*)d_in[5];

    // out = pooled[B,K,D] ++ masked_probs[B,S] ++ shortened[B] ++ nb[B] ++ actual_lens[B]
    int K = (out_size - BDIM * SDIM - 3 * BDIM) / (BDIM * DDIM);
    if (K < 0) K = 0;
    int max_segments = K > 1 ? K : 1;

    float* out        = (float*)d_out;
    float* out_pooled = out;
    float* out_mask   = out_pooled + (size_t)BDIM * K * DDIM;
    float* out_sh     = out_mask + (size_t)BDIM * SDIM;
    float* out_nb     = out_sh + BDIM;
    float* out_al     = out_nb + BDIM;

    float* logits_ws = (float*)d_ws;                 // MTOT floats
    float* hard_ws   = logits_ws + MTOT;             // MTOT floats
    int*   nb_ws     = (int*)(hard_ws + MTOT);       // BDIM ints

    zero_kernel<<<(MTOT + 255) / 256, 256, 0, stream>>>(logits_ws, nb_ws);

    dim3 ggrid(MTOT / BM, DDIM / BN, 1);             // 128 x 6
    gemm_gelu_dot<<<ggrid, 256, 0, stream>>>(hidden, W1, b1, W2, logits_ws);

    boundary_kernel<<<(MTOT + 255) / 256, 256, 0, stream>>>(
        logits_ws, lengths, b2, out_mask, hard_ws, nb_ws);

    finalize_kernel<<<1, 32, 0, stream>>>(nb_ws, lengths, out_sh, out_nb, out_al,
                                          max_segments);

    dim3 pgrid(BDIM, DDIM / 128, 1);                 // 8 x 6
    pool_kernel<<<pgrid, 128, 0, stream>>>(hidden, hard_ws, lengths, out_pooled, K);
}